// Model_24043226923191
// MI455X (gfx1250) — compile-verified
//
#include <hip/hip_runtime.h>

typedef __attribute__((ext_vector_type(2))) float v2f;
typedef __attribute__((ext_vector_type(8))) float v8f;
typedef __attribute__((ext_vector_type(4))) unsigned int u32x4;
typedef __attribute__((ext_vector_type(8))) int i32x8;
typedef __attribute__((ext_vector_type(4))) int i32x4;

#define NEG_INF (-3.4e38f)

// ---------------------------------------------------------------------------
// prep: X0 = concat(x, fx) -> (B*N, 8) padded with a zero channel
// ---------------------------------------------------------------------------
__global__ void prep_x0_kernel(const float* __restrict__ x, const float* __restrict__ fx,
                               float* __restrict__ X0, int NP) {
  int p = blockIdx.x * blockDim.x + threadIdx.x;
  if (p >= NP) return;
  X0[p * 8 + 0] = x[p * 3 + 0];
  X0[p * 8 + 1] = x[p * 3 + 1];
  X0[p * 8 + 2] = x[p * 3 + 2];
  X0[p * 8 + 3] = fx[p * 4 + 0];
  X0[p * 8 + 4] = fx[p * 4 + 1];
  X0[p * 8 + 5] = fx[p * 4 + 2];
  X0[p * 8 + 6] = fx[p * 4 + 3];
  X0[p * 8 + 7] = 0.0f;
}

// ---------------------------------------------------------------------------
// prep: Wcat (2O x Cpad). rows 0..O-1 = W1, rows O..2O-1 = W2 - W1  (zero pad)
// source w is (O x 2C) row-major: cols 0..C-1 = W1 (nbr-ctr), C..2C-1 = W2 (ctr)
// ---------------------------------------------------------------------------
__global__ void prep_wcat_kernel(const float* __restrict__ w, int O, int C, int Cpad,
                                 float* __restrict__ wcat) {
  int i = blockIdx.x * blockDim.x + threadIdx.x;
  int total = 2 * O * Cpad;
  if (i >= total) return;
  int c = i % Cpad;
  int r = i / Cpad;
  float v = 0.0f;
  if (c < C) {
    if (r < O) {
      v = w[r * (2 * C) + c];
    } else {
      int o = r - O;
      v = w[o * (2 * C) + C + c] - w[o * (2 * C) + c];
    }
  }
  wcat[i] = v;
}

// ---------------------------------------------------------------------------
// xx[p] = sum_c X[p][c]^2   (strided input)
// ---------------------------------------------------------------------------
__global__ void xx_kernel(const float* __restrict__ X, int stride, int C,
                          float* __restrict__ xx, int NP) {
  int p = blockIdx.x * blockDim.x + threadIdx.x;
  if (p >= NP) return;
  const float* row = X + (size_t)p * stride;
  float s = 0.0f;
  for (int c = 0; c < C; ++c) { float v = row[c]; s += v * v; }
  xx[p] = s;
}

// ---------------------------------------------------------------------------
// WMMA helpers: f32 16x16x4, wave32.
// A layout: lane&15 = M, (lane>=16 -> K base 2); D: VGPR v = rows {v, v+8}
// ---------------------------------------------------------------------------
__device__ __forceinline__ v8f wmma4(v2f a, v2f b, v8f c) {
  return __builtin_amdgcn_wmma_f32_16x16x4_f32(false, a, false, b, (short)0, c,
                                               false, false);
}

// ---------------------------------------------------------------------------
// Gram panel via WMMA, 32x32 per wave (2x2 register tile, A/B reuse):
//   D[r - R0][m] = 2 * (x_r . x_m) - xx[r] - xx[m]  for r in [R0, R0+512)
// Block = 128 threads (4 waves); wave w covers col tile blockIdx.x*4 + w.
// ---------------------------------------------------------------------------
__global__ void gram_kernel(const float* __restrict__ X, int stride, int C,
                            const float* __restrict__ xx, int R0,
                            float* __restrict__ D, int N) {
  int wave = threadIdx.x >> 5;
  int lane = threadIdx.x & 31;
  int rt = blockIdx.y;                    // 32-row tile within panel
  int ct = blockIdx.x * 4 + wave;         // 32-col tile (global)
  int m15 = lane & 15;
  int koff = (lane >> 4) << 1;            // 0 or 2

  int r0g = R0 + rt * 32;                 // global row base
  int c0g = ct * 32;                      // global col base
  const float* ap0 = X + (size_t)(r0g + m15) * stride + koff;
  const float* ap1 = ap0 + (size_t)16 * stride;
  const float* bp0 = X + (size_t)(c0g + m15) * stride + koff;
  const float* bp1 = bp0 + (size_t)16 * stride;

  v8f acc00 = {}, acc01 = {}, acc10 = {}, acc11 = {};
  for (int kc = 0; kc < C; kc += 4) {
    v2f a0, a1, b0, b1;
    a0.x = ap0[kc]; a0.y = ap0[kc + 1];
    a1.x = ap1[kc]; a1.y = ap1[kc + 1];
    b0.x = bp0[kc]; b0.y = bp0[kc + 1];
    b1.x = bp1[kc]; b1.y = bp1[kc + 1];
    acc00 = wmma4(a0, b0, acc00);
    acc01 = wmma4(a0, b1, acc01);
    acc10 = wmma4(a1, b0, acc10);
    acc11 = wmma4(a1, b1, acc11);
  }

  int dn = lane & 15;
  int mb = (lane < 16) ? 0 : 8;
  int rloc0 = rt * 32;                    // panel-local row base
  union { v8f v; float f[8]; } u;

#define GRAM_STORE(ACC, RADD, CADD)                                            \
  u.v = (ACC);                                                                 \
  {                                                                            \
    int col = c0g + (CADD) + dn;                                               \
    float xxc = xx[col];                                                       \
    for (int i = 0; i < 8; ++i) {                                              \
      int m = mb + i + (RADD);                                                 \
      float d = 2.0f * u.f[i] - xx[r0g + m] - xxc;                             \
      D[(size_t)(rloc0 + m) * N + col] = d;                                    \
    }                                                                          \
  }
  GRAM_STORE(acc00, 0, 0)
  GRAM_STORE(acc01, 0, 16)
  GRAM_STORE(acc10, 16, 0)
  GRAM_STORE(acc11, 16, 16)
#undef GRAM_STORE
}

// ---------------------------------------------------------------------------
// top-k(40) per row; row staged in LDS via the Tensor Data Mover; iterative
// argmax (ties -> lowest idx, matching lax.top_k stability)
// ---------------------------------------------------------------------------
__global__ void topk_kernel(const float* __restrict__ D, int* __restrict__ idxOut,
                            int N, int K) {
  __shared__ float vals[4096];
  __shared__ float rv[256];
  __shared__ int ri[256];
  const float* row = D + (size_t)blockIdx.x * N;
  int t = threadIdx.x;

#if __has_builtin(__builtin_amdgcn_tensor_load_to_lds) && \
    __has_builtin(__builtin_amdgcn_s_wait_tensorcnt)
  // Tensor Data Mover: DMA the 16KB row straight into LDS (wave 0 issues).
  if (t < 32) {
    unsigned long long ga = (unsigned long long)(uintptr_t)row;
    unsigned lds_off = (unsigned)(uintptr_t)(&vals[0]);
    u32x4 g0;
    g0.x = 1u;                                        // count=1, user mode
    g0.y = lds_off;                                   // lds_addr
    g0.z = (unsigned)(ga & 0xffffffffu);              // global_addr[31:0]
    g0.w = (unsigned)((ga >> 32) & 0x1ffffffu) | (2u << 30);  // [56:32] | type=2
    i32x8 g1;
    g1[0] = (int)(2u << 16);          // data_size=4B, no multicast, no flags
    g1[1] = (int)(4096u << 16);       // tensor_dim0[15:0] (=4096) in bits 63:48
    g1[2] = (int)(1u << 16);          // tensor_dim0 hi=0 | tensor_dim1 lo = 1
    g1[3] = (int)(4096u << 16);       // tensor_dim1 hi=0 | tile_dim0 = 4096
    g1[4] = 0;                        // tile_dim1 = 0 (unused), tile_dim2 = 0
    g1[5] = 4096;                     // tensor_dim0_stride[31:0]
    g1[6] = 0;
    g1[7] = 0;
    i32x4 gz4 = {};
    i32x8 gz8 = {};
    __builtin_amdgcn_tensor_load_to_lds(g0, g1, gz4, gz4, gz8, 0);
    __builtin_amdgcn_s_wait_tensorcnt(0);
  }
  __syncthreads();
#else
  for (int i = t; i < N; i += 256) vals[i] = row[i];
  __syncthreads();
#endif

  for (int k = 0; k < K; ++k) {
    float best = NEG_INF;
    int bi = N;
    for (int i = t; i < N; i += 256) {
      float v = vals[i];
      if (v > best) { best = v; bi = i; }   // first hit => lowest index on ties
    }
    rv[t] = best; ri[t] = bi;
    __syncthreads();
    for (int s = 128; s > 0; s >>= 1) {
      if (t < s) {
        float ov = rv[t + s]; int oi = ri[t + s];
        if (ov > rv[t] || (ov == rv[t] && oi < ri[t])) { rv[t] = ov; ri[t] = oi; }
      }
      __syncthreads();
    }
    if (t == 0) {
      idxOut[(size_t)blockIdx.x * K + k] = ri[0];
      vals[ri[0]] = NEG_INF;
    }
    __syncthreads();
  }
}

// ---------------------------------------------------------------------------
// Generic row-major GEMM via WMMA, 32x32 per wave (2x2 register tile):
//   Out[p][o] = sum_c X[p][c] * W[o][c]
// mode 1: += bias[o], relu.  Block = 128 threads (4 waves):
//   out tile (32 wide) = blockIdx.x, point tile (32 high) = blockIdx.y*4 + wave
// ---------------------------------------------------------------------------
__global__ void gemm_kernel(const float* __restrict__ X, int strideX, int C,
                            const float* __restrict__ W, int strideW,
                            float* __restrict__ Out, int strideOut,
                            const float* __restrict__ bias, int mode) {
  int wave = threadIdx.x >> 5;
  int lane = threadIdx.x & 31;
  int ot = blockIdx.x;
  int pt = blockIdx.y * 4 + wave;
  int m15 = lane & 15;
  int koff = (lane >> 4) << 1;

  int p0 = pt * 32;
  int o0 = ot * 32;
  const float* ap0 = X + (size_t)(p0 + m15) * strideX + koff;
  const float* ap1 = ap0 + (size_t)16 * strideX;
  const float* bp0 = W + (size_t)(o0 + m15) * strideW + koff;
  const float* bp1 = bp0 + (size_t)16 * strideW;

  v8f acc00 = {}, acc01 = {}, acc10 = {}, acc11 = {};
  for (int kc = 0; kc < C; kc += 4) {
    v2f a0, a1, b0, b1;
    a0.x = ap0[kc]; a0.y = ap0[kc + 1];
    a1.x = ap1[kc]; a1.y = ap1[kc + 1];
    b0.x = bp0[kc]; b0.y = bp0[kc + 1];
    b1.x = bp1[kc]; b1.y = bp1[kc + 1];
    acc00 = wmma4(a0, b0, acc00);
    acc01 = wmma4(a0, b1, acc01);
    acc10 = wmma4(a1, b0, acc10);
    acc11 = wmma4(a1, b1, acc11);
  }

  int dn = lane & 15;
  int mb = (lane < 16) ? 0 : 8;
  union { v8f v; float f[8]; } u;

#define GEMM_STORE(ACC, RADD, CADD)                                            \
  u.v = (ACC);                                                                 \
  {                                                                            \
    int ocol = o0 + (CADD) + dn;                                               \
    float bv = (mode == 1) ? bias[ocol] : 0.0f;                                \
    for (int i = 0; i < 8; ++i) {                                              \
      int m = mb + i + (RADD);                                                 \
      float val = u.f[i];                                                      \
      if (mode == 1) { val += bv; val = fmaxf(val, 0.0f); }                    \
      Out[(size_t)(p0 + m) * strideOut + ocol] = val;                          \
    }                                                                          \
  }
  GEMM_STORE(acc00, 0, 0)
  GEMM_STORE(acc01, 0, 16)
  GEMM_STORE(acc10, 16, 0)
  GEMM_STORE(acc11, 16, 16)
#undef GEMM_STORE
}

// ---------------------------------------------------------------------------
// EdgeConv gather: z(p,k,o) = T[nbr][o] + T[p][O+o]; track max_k, sum, sumsq
// Block per point, thread per channel. Tables live in L2 (<=16MB << 192MB).
// ---------------------------------------------------------------------------
__global__ void gather_kernel(const float* __restrict__ T, int twoO, int O,
                              const int* __restrict__ idx,
                              float* __restrict__ zmax,
                              float* __restrict__ csum, float* __restrict__ csumsq,
                              int N, int K) {
  int p = blockIdx.x;
  int t = threadIdx.x;
  int base = p & ~(N - 1);            // batch base (N is a power of two)
  float cen = T[(size_t)p * twoO + O + t];
  float mx = NEG_INF, s = 0.0f, s2 = 0.0f;
  for (int k = 0; k < K; ++k) {
    int j = base + idx[(size_t)p * K + k];
    float z = T[(size_t)j * twoO + t] + cen;
    mx = fmaxf(mx, z);
    s += z;
    s2 += z * z;
  }
  zmax[(size_t)p * O + t] = mx;
  atomicAdd(&csum[t], s);
  atomicAdd(&csumsq[t], s2);
}

// ---------------------------------------------------------------------------
// BN(train) + LeakyReLU applied to the per-point max (monotone => commutes)
// ---------------------------------------------------------------------------
__global__ void bn_max_kernel(const float* __restrict__ zmax, int O,
                              const float* __restrict__ g, const float* __restrict__ bb,
                              const float* __restrict__ csum, const float* __restrict__ csumsq,
                              float invcnt, float* __restrict__ out, int strideOut) {
  int p = blockIdx.x;
  int t = threadIdx.x;
  float m = csum[t] * invcnt;
  float var = csumsq[t] * invcnt - m * m;
  float sc = g[t] * rsqrtf(var + 1e-5f);
  float v = (zmax[(size_t)p * O + t] - m) * sc + bb[t];
  out[(size_t)p * strideOut + t] = (v > 0.0f) ? v : 0.2f * v;
}

// ---------------------------------------------------------------------------
// per-channel sum / sumsq over rows (for conv1d BN)
// ---------------------------------------------------------------------------
__global__ void colsum_kernel(const float* __restrict__ Z, int O, int rowsPerBlock,
                              float* __restrict__ csum, float* __restrict__ csumsq) {
  int c = blockIdx.x * blockDim.x + threadIdx.x;
  int r0 = blockIdx.y * rowsPerBlock;
  float s = 0.0f, s2 = 0.0f;
  for (int r = r0; r < r0 + rowsPerBlock; ++r) {
    float v = Z[(size_t)r * O + c];
    s += v;
    s2 += v * v;
  }
  atomicAdd(&csum[c], s);
  atomicAdd(&csumsq[c], s2);
}

__global__ void bn_apply_kernel(float* __restrict__ Z, int O,
                                const float* __restrict__ g, const float* __restrict__ bb,
                                const float* __restrict__ csum, const float* __restrict__ csumsq,
                                float invcnt) {
  size_t p = blockIdx.x;
  for (int c = threadIdx.x; c < O; c += blockDim.x) {
    float m = csum[c] * invcnt;
    float var = csumsq[c] * invcnt - m * m;
    float sc = g[c] * rsqrtf(var + 1e-5f);
    float v = (Z[p * O + c] - m) * sc + bb[c];
    Z[p * O + c] = (v > 0.0f) ? v : 0.2f * v;
  }
}

// ---------------------------------------------------------------------------
// out[p] = h1[p] . wp2 + bp2  (64-wide dot, trivial)
// ---------------------------------------------------------------------------
__global__ void final_kernel(const float* __restrict__ h1, const float* __restrict__ wp2,
                             const float* __restrict__ bp2, float* __restrict__ out, int NP) {
  int p = blockIdx.x * blockDim.x + threadIdx.x;
  if (p >= NP) return;
  float s = bp2[0];
  const float* row = h1 + (size_t)p * 64;
  for (int c = 0; c < 64; ++c) s += row[c] * wp2[c];
  out[p] = s;
}

// ---------------------------------------------------------------------------
extern "C" void kernel_launch(void* const* d_in, const int* in_sizes, int n_in,
                              void* d_out, int out_size, void* d_ws, size_t ws_size,
                              hipStream_t stream) {
  (void)in_sizes; (void)n_in; (void)out_size; (void)ws_size;
  const int B = 2, N = 4096, K = 40, NP = B * N;

  const float* x   = (const float*)d_in[0];
  const float* fx  = (const float*)d_in[1];
  const float* w1  = (const float*)d_in[2];
  const float* g1  = (const float*)d_in[3];
  const float* b1  = (const float*)d_in[4];
  const float* w2  = (const float*)d_in[5];
  const float* g2  = (const float*)d_in[6];
  const float* b2  = (const float*)d_in[7];
  const float* w3  = (const float*)d_in[8];
  const float* g3  = (const float*)d_in[9];
  const float* b3  = (const float*)d_in[10];
  const float* w4  = (const float*)d_in[11];
  const float* g4  = (const float*)d_in[12];
  const float* b4  = (const float*)d_in[13];
  const float* w5  = (const float*)d_in[14];
  const float* g5  = (const float*)d_in[15];
  const float* b5  = (const float*)d_in[16];
  const float* wp1 = (const float*)d_in[17];
  const float* bp1 = (const float*)d_in[18];
  const float* wp2 = (const float*)d_in[19];
  const float* bp2 = (const float*)d_in[20];
  float* out = (float*)d_out;

  char* wsb = (char*)d_ws;
  size_t off = 0;
  auto alloc = [&](size_t bytes) -> void* {
    void* p = wsb + off;
    off += (bytes + 255) & ~(size_t)255;
    return p;
  };
  float* X0     = (float*)alloc((size_t)NP * 8 * 4);
  float* cat    = (float*)alloc((size_t)NP * 576 * 4);
  float* T      = (float*)alloc((size_t)NP * 512 * 4);
  float* zmax   = (float*)alloc((size_t)NP * 256 * 4);
  float* Z      = (float*)alloc((size_t)NP * 1024 * 4);
  float* h1     = (float*)alloc((size_t)NP * 64 * 4);
  float* dist   = (float*)alloc((size_t)512 * N * 4);
  int*   idx    = (int*)alloc((size_t)NP * K * 4);
  float* wcat   = (float*)alloc((size_t)512 * 256 * 4);
  float* xx     = (float*)alloc((size_t)NP * 4);
  float* csum   = (float*)alloc(1024 * 4);
  float* csumsq = (float*)alloc(1024 * 4);

  prep_x0_kernel<<<NP / 256, 256, 0, stream>>>(x, fx, X0, NP);

  struct L { int C, Cpad, O, choff; const float *w, *g, *bb; };
  L Ls[4] = {
    {  7,   8,  64,   0, w1, g1, b1},
    { 64,  64, 128,  64, w2, g2, b2},
    {128, 128, 128, 192, w3, g3, b3},
    {128, 128, 256, 320, w4, g4, b4},
  };

  const float* Xin = X0;
  int sIn = 8;
  for (int l = 0; l < 4; ++l) {
    const L& LL = Ls[l];
    int twoO = 2 * LL.O;

    prep_wcat_kernel<<<(2 * LL.O * LL.Cpad + 255) / 256, 256, 0, stream>>>(
        LL.w, LL.O, LL.C, LL.Cpad, wcat);
    xx_kernel<<<NP / 256, 256, 0, stream>>>(Xin, sIn, LL.Cpad, xx, NP);

    // kNN: WMMA Gram panels (512 rows) -> top-40 per row
    for (int b = 0; b < B; ++b) {
      const float* Xb = Xin + (size_t)b * N * sIn;
      const float* xxb = xx + (size_t)b * N;
      for (int p0 = 0; p0 < N; p0 += 512) {
        gram_kernel<<<dim3(N / 128, 16), 128, 0, stream>>>(Xb, sIn, LL.Cpad, xxb, p0,
                                                           dist, N);
        topk_kernel<<<512, 256, 0, stream>>>(dist, idx + ((size_t)b * N + p0) * K,
                                             N, K);
      }
    }

    // T = X . [W1 ; W2-W1]^T  (per-point tables, WMMA)
    gemm_kernel<<<dim3(twoO / 32, NP / 128), 128, 0, stream>>>(
        Xin, sIn, LL.Cpad, wcat, LL.Cpad, T, twoO, nullptr, 0);

    (void)hipMemsetAsync(csum, 0, 1024 * 4, stream);
    (void)hipMemsetAsync(csumsq, 0, 1024 * 4, stream);
    gather_kernel<<<NP, LL.O, 0, stream>>>(T, twoO, LL.O, idx, zmax, csum, csumsq,
                                           N, K);
    bn_max_kernel<<<NP, LL.O, 0, stream>>>(zmax, LL.O, LL.g, LL.bb, csum, csumsq,
                                           1.0f / ((float)NP * (float)K),
                                           cat + LL.choff, 576);
    Xin = cat + LL.choff;
    sIn = 576;
  }

  // conv1d block: Z = cat . w5^T (WMMA), BN(train) + LeakyReLU over (B,N)
  gemm_kernel<<<dim3(1024 / 32, NP / 128), 128, 0, stream>>>(cat, 576, 576, w5, 576,
                                                             Z, 1024, nullptr, 0);
  (void)hipMemsetAsync(csum, 0, 1024 * 4, stream);
  (void)hipMemsetAsync(csumsq, 0, 1024 * 4, stream);
  colsum_kernel<<<dim3(1024 / 256, 16), 256, 0, stream>>>(Z, 1024, NP / 16, csum,
                                                          csumsq);
  bn_apply_kernel<<<NP, 256, 0, stream>>>(Z, 1024, g5, b5, csum, csumsq,
                                          1.0f / (float)NP);

  // MLP head: h1 = relu(Z . wp1^T + bp1) (WMMA), out = h1 . wp2 + bp2
  gemm_kernel<<<dim3(64 / 32, NP / 128), 128, 0, stream>>>(Z, 1024, 1024, wp1, 1024,
                                                           h1, 64, bp1, 1);
  final_kernel<<<NP / 256, 256, 0, stream>>>(h1, wp2, bp2, out, NP);
}